// ACECalculator_87436944212794
// MI455X (gfx1250) — compile-verified
//
#include <hip/hip_runtime.h>

#define DEV_INLINE __device__ __forceinline__

// ---------------------------------------------------------------------------
// CDNA5 async global->LDS staging (ASYNCcnt-tracked), guarded so the WMMA
// path survives on toolchains that don't declare the async builtins.
// Builtin signature (from hipcc diagnostic): first param '__device__ int *'
// i.e. (addrspace(1) int* gsrc, addrspace(3) int* ldst, imm offset, imm cpol).
// ---------------------------------------------------------------------------
#if defined(__has_builtin)
#if __has_builtin(__builtin_amdgcn_global_load_async_to_lds_b32) && \
    __has_builtin(__builtin_amdgcn_s_wait_asynccnt)
#define USE_ASYNC_LDS 1
#endif
#endif
#ifndef USE_ASYNC_LDS
#define USE_ASYNC_LDS 0
#endif

namespace {

constexpr int NRAD   = 16;     // radial channels per l
constexpr int NATOMS = 1000;   // n_i
constexpr int P2C    = 128;
constexpr int P3C    = 256;
constexpr int NP2C   = 2;
constexpr int NP3C   = 4;
constexpr int ATILE  = 16;     // atoms per tile = WMMA N dimension
constexpr int QBLK   = 64;     // q values per block
constexpr int QPW    = 8;      // q values per wave (QBLK / 8 waves)

typedef __attribute__((ext_vector_type(2))) float v2f;
typedef __attribute__((ext_vector_type(8))) float v8f;

typedef __attribute__((address_space(1))) int gas_int;   // global (device) int
typedef __attribute__((address_space(3))) int las_int;   // LDS (shared) int

DEV_INLINE void stage_b32(void* lds_dst, const void* gsrc) {
#if USE_ASYNC_LDS
  __builtin_amdgcn_global_load_async_to_lds_b32(
      (gas_int*)gsrc, (las_int*)lds_dst, /*offset=*/0, /*cpol=*/0);
#else
  *(unsigned*)lds_dst = *(const unsigned*)gsrc;
#endif
}

DEV_INLINE void stage_fence() {
#if USE_ASYNC_LDS
  __builtin_amdgcn_s_wait_asynccnt(0);   // s_wait_asynccnt 0
#endif
  __syncthreads();
}

// LDS layout for the per-tile LE cache: LE[l][(r*ml+m)] x 16 atoms
//   l=0 : rm in [0,16)  -> base 0,    256 floats
//   l=1 : rm in [0,48)  -> base 256,  768 floats
//   l=2 : rm in [0,80)  -> base 1024, 1280 floats
constexpr int LDS_LE_TOTAL = 2304;
constexpr int ldsBaseForM(int M) { return (M == 1) ? 0 : ((M == 3) ? 256 : 1024); }

// A3 element g = (m1*MB + m2)*MC + m3 built from register-cached LE rows.
// g is a compile-time constant after full unroll -> static register indexing.
template<int MA, int MB, int MC>
DEV_INLINE float a3val(int g, const float* va, const float* vb, const float* vc) {
  constexpr int F3 = MA * MB * MC;
  if (g >= F3) return 0.0f;           // folds at compile time (zero K padding)
  const int m3 = g % MC;
  const int t  = g / MC;
  const int m2 = t % MB;
  const int m1 = t / MB;
  return va[m1] * vb[m2] * vc[m3];
}

// ---------------------------------------------------------------------------
// nu=3 kernel: out3[path, q, n] = mult3[q] * sum_g cg3[path, g] * A3[g, n]
// One block: 16-atom tile x 64 q values. Each wave handles 8 q values.
// Projection via V_WMMA_F32_16X16X4_F32:
//   D[16,16] (+)= A[16,4] x B[4,16], M=paths (4 valid), N=atoms, K=g chunks.
// A-operand (cg3) chunks are q-invariant -> hoisted into registers per wave.
// ---------------------------------------------------------------------------
template<int MA, int MB, int MC>
__global__ __launch_bounds__(256)
void ace3_wmma_kernel(const float* __restrict__ LE0,
                      const float* __restrict__ LE1,
                      const float* __restrict__ LE2,
                      const int*   __restrict__ i0arr,   // combine_idx2[t2][0], [P2] -> [0,NRAD)
                      const int*   __restrict__ i1arr,   // combine_idx2[t2][1]
                      const int*   __restrict__ j0arr,   // combine_idx3[t3][0], [P3] -> [0,P2)
                      const int*   __restrict__ j1arr,   // combine_idx3[t3][1], [P3] -> [0,NRAD)
                      const float* __restrict__ mult3,   // [P3]
                      const float* __restrict__ cg3,     // [NP3, F3]
                      float*       __restrict__ out)     // [NP3*P3, NATOMS]
{
  constexpr int F3  = MA * MB * MC;
  constexpr int NCH = (F3 + 3) / 4;    // K chunks of 4

  __shared__ float sLE[LDS_LE_TOTAL];
  __shared__ float sCG[NP3C * 125];
  __shared__ int   sJ0[P3C];
  __shared__ int   sJ1[P3C];
  __shared__ float sM3[P3C];
  __shared__ int   sI0[P2C];
  __shared__ int   sI1[P2C];

  const int tid = threadIdx.x;
  const int n0  = blockIdx.x * ATILE;

  // ---- Stage everything the block touches into LDS (async on CDNA5).
  for (int e = tid; e < LDS_LE_TOTAL; e += 256) {
    const float* src;
    int base;
    if (e < 256)       { base = 0;    src = LE0; }
    else if (e < 1024) { base = 256;  src = LE1; }
    else               { base = 1024; src = LE2; }
    const int off = e - base;
    const int rm  = off >> 4;                     // r*ml + m
    int n = n0 + (off & 15);
    if (n > NATOMS - 1) n = NATOMS - 1;           // clamp; tail lanes masked at store
    stage_b32(&sLE[e], &src[rm * NATOMS + n]);
  }
  for (int e = tid; e < NP3C * F3; e += 256) stage_b32(&sCG[e], &cg3[e]);
  if (tid < P3C) {
    stage_b32(&sJ0[tid], &j0arr[tid]);
    stage_b32(&sJ1[tid], &j1arr[tid]);
    stage_b32(&sM3[tid], &mult3[tid]);
  }
  if (tid < P2C) {
    stage_b32(&sI0[tid], &i0arr[tid]);
    stage_b32(&sI1[tid], &i1arr[tid]);
  }
  stage_fence();

  const int  lane = tid & 31;
  const int  wave = tid >> 5;
  const int  atom = lane & 15;        // WMMA N index (atom) / M index (path) role
  const int  half = lane >> 4;        // K-half select per ISA operand layout
  const int  n    = n0 + atom;
  const bool nOK  = (n < NATOMS);
  const int  qbase = blockIdx.y * QBLK + wave * QPW;

  // ---- Hoist the q-invariant A operand (cg3) into registers, ISA layout:
  // lanes 0-15 carry K=kb,kb+1 ; lanes 16-31 carry K=kb+2,kb+3 ; M=path=lane&15.
  v2f Areg[NCH];
#pragma unroll
  for (int ch = 0; ch < NCH; ++ch) {
    const int kb   = 4 * ch;
    const int path = atom;
    const int pcl  = (path < 4) ? path : 3;
    const int k0   = kb + 2 * half;
    const int k0c  = (k0 < F3) ? k0 : (F3 - 1);
    const int k1c  = (k0 + 1 < F3) ? (k0 + 1) : (F3 - 1);
    float a0 = sCG[pcl * F3 + k0c];
    float a1 = sCG[pcl * F3 + k1c];
    a0 = (path < 4 && k0     < F3) ? a0 : 0.0f;
    a1 = (path < 4 && k0 + 1 < F3) ? a1 : 0.0f;
    Areg[ch].x = a0;
    Areg[ch].y = a1;
  }

  for (int qi = 0; qi < QPW; ++qi) {
    const int   q  = qbase + qi;
    const int   p  = sJ0[q];
    const int   r  = sJ1[q];
    const int   i0 = sI0[p];
    const int   i1 = sI1[p];
    const float ml = sM3[q];

    // Register-cache this atom's three LE rows (<= 5+5+5 floats).
    float va[MA], vb[MB], vc[MC];
#pragma unroll
    for (int m = 0; m < MA; ++m) va[m] = sLE[ldsBaseForM(MA) + (i0 * MA + m) * 16 + atom];
#pragma unroll
    for (int m = 0; m < MB; ++m) vb[m] = sLE[ldsBaseForM(MB) + (i1 * MB + m) * 16 + atom];
#pragma unroll
    for (int m = 0; m < MC; ++m) vc[m] = sLE[ldsBaseForM(MC) + (r  * MC + m) * 16 + atom];

    v8f acc = {0.f, 0.f, 0.f, 0.f, 0.f, 0.f, 0.f, 0.f};

#pragma unroll
    for (int ch = 0; ch < NCH; ++ch) {
      const int kb = 4 * ch;
      // A3 values for this atom at K = kb..kb+3 (compile-time decode, CSE'd).
      const float p0 = a3val<MA, MB, MC>(kb + 0, va, vb, vc);
      const float p1 = a3val<MA, MB, MC>(kb + 1, va, vb, vc);
      const float p2 = a3val<MA, MB, MC>(kb + 2, va, vb, vc);
      const float p3 = a3val<MA, MB, MC>(kb + 3, va, vb, vc);

      v2f B;
      B.x = half ? p2 : p0;   // N=atom, mirrored K halves
      B.y = half ? p3 : p1;

      acc = __builtin_amdgcn_wmma_f32_16x16x4_f32(
          /*neg_a=*/false, Areg[ch], /*neg_b=*/false, B,
          /*c_mod=*/(short)0, acc, /*reuse_a=*/false, /*reuse_b=*/false);
    }

    // D layout: VGPR rr, lanes 0-15 => M=rr (path), N=lane (atom). Paths 0..3 valid.
    if (lane < 16 && nOK) {
#pragma unroll
      for (int rr = 0; rr < NP3C; ++rr) {
        out[(size_t)(rr * P3C + q) * NATOMS + n] = ml * acc[rr];
      }
    }
  }
}

// ---------------------------------------------------------------------------
// nu=2 kernel (7% of output): out2[path, p, n] = mult2[p] * sum_f cg2[path,f]
//                                               * LE[i0[p],m1,n]*LE[i1[p],m2,n]
// ---------------------------------------------------------------------------
template<int M>
__global__ __launch_bounds__(256)
void ace2_kernel(const float* __restrict__ LE,
                 const int*   __restrict__ i0arr,
                 const int*   __restrict__ i1arr,
                 const float* __restrict__ mult2,
                 const float* __restrict__ cg2,     // [NP2, M*M]
                 float*       __restrict__ out)     // [NP2*P2, NATOMS]
{
  const int p = blockIdx.x;
  const int n = blockIdx.y * 256 + threadIdx.x;
  if (n >= NATOMS) return;

  const int i0 = i0arr[p];
  const int i1 = i1arr[p];

  float va[M], vb[M];
#pragma unroll
  for (int m = 0; m < M; ++m) {
    va[m] = LE[(i0 * M + m) * NATOMS + n];
    vb[m] = LE[(i1 * M + m) * NATOMS + n];
  }

  float s0 = 0.f, s1 = 0.f;
#pragma unroll
  for (int m1 = 0; m1 < M; ++m1) {
#pragma unroll
    for (int m2 = 0; m2 < M; ++m2) {
      const float pr = va[m1] * vb[m2];
      const int   f  = m1 * M + m2;
      s0 = fmaf(cg2[f],         pr, s0);
      s1 = fmaf(cg2[M * M + f], pr, s1);
    }
  }
  const float ml = mult2[p];
  out[(size_t)(0 * P2C + p) * NATOMS + n] = ml * s0;
  out[(size_t)(1 * P2C + p) * NATOMS + n] = ml * s1;
}

} // anonymous namespace

// ---------------------------------------------------------------------------
// Input layout (setup_inputs insertion order, pytree-flattened):
//   [0..2]   LE_1[l], l=0,1,2                         float32
//   [3..8]   mult2[t2], SORTED2 order                 float32[128]
//   [9..18]  mult3[t3], SORTED3 order                 float32[256]
//   [19..21] cg2[(l,l)], l=0,1,2                      float32[2, (2l+1)^2]
//   [22..31] cg3[t3], SORTED3 order                   float32[4, F3]
//   [32..43] combine_idx2: per t2 (idx0, idx1)        int32[128] x2
//   [44..63] combine_idx3: per t3 (idx0, idx1)        int32[256] x2
// Output: 3 x [2*128*1000] (t2 in sorted cg2 order) then
//         10 x [4*256*1000] (t3 in SORTED3 order).
// ---------------------------------------------------------------------------
extern "C" void kernel_launch(void* const* d_in, const int* in_sizes, int n_in,
                              void* d_out, int out_size, void* d_ws, size_t ws_size,
                              hipStream_t stream) {
  (void)in_sizes; (void)n_in; (void)out_size; (void)d_ws; (void)ws_size;

  const float* LE[3];
  for (int i = 0; i < 3; ++i) LE[i] = (const float*)d_in[i];
  const float* mult2[6];
  for (int i = 0; i < 6; ++i) mult2[i] = (const float*)d_in[3 + i];
  const float* mult3[10];
  for (int i = 0; i < 10; ++i) mult3[i] = (const float*)d_in[9 + i];
  const float* cg2[3];
  for (int i = 0; i < 3; ++i) cg2[i] = (const float*)d_in[19 + i];
  const float* cg3[10];
  for (int i = 0; i < 10; ++i) cg3[i] = (const float*)d_in[22 + i];
  const int* cidx2[12];
  for (int i = 0; i < 12; ++i) cidx2[i] = (const int*)d_in[32 + i];
  const int* cidx3[20];
  for (int i = 0; i < 20; ++i) cidx3[i] = (const int*)d_in[44 + i];

  float* out = (float*)d_out;

  // ---- nu=2 outputs: t2 = (0,0), (1,1), (2,2) at SORTED2 positions 0, 3, 5
  const size_t O2 = (size_t)NP2C * P2C * NATOMS;
  const dim3 g2(P2C, (NATOMS + 255) / 256);
  ace2_kernel<1><<<g2, 256, 0, stream>>>(LE[0], cidx2[0],  cidx2[1],  mult2[0], cg2[0], out + 0 * O2);
  ace2_kernel<3><<<g2, 256, 0, stream>>>(LE[1], cidx2[6],  cidx2[7],  mult2[3], cg2[1], out + 1 * O2);
  ace2_kernel<5><<<g2, 256, 0, stream>>>(LE[2], cidx2[10], cidx2[11], mult2[5], cg2[2], out + 2 * O2);

  // ---- nu=3 outputs: SORTED3 order; t2-index of (a,b) prefix per tuple
  static const int T3T2[10] = {0, 0, 0, 1, 1, 2, 3, 3, 4, 5};
  const size_t base3 = 3 * O2;
  const size_t O3 = (size_t)NP3C * P3C * NATOMS;
  const dim3 g3((NATOMS + ATILE - 1) / ATILE, P3C / QBLK);

#define LAUNCH3(K, MA_, MB_, MC_)                                               \
  ace3_wmma_kernel<MA_, MB_, MC_><<<g3, 256, 0, stream>>>(                      \
      LE[0], LE[1], LE[2],                                                      \
      cidx2[2 * T3T2[K]], cidx2[2 * T3T2[K] + 1],                               \
      cidx3[2 * (K)], cidx3[2 * (K) + 1],                                       \
      mult3[K], cg3[K], out + base3 + (size_t)(K) * O3);

  LAUNCH3(0, 1, 1, 1)   // (0,0,0)
  LAUNCH3(1, 1, 1, 3)   // (0,0,1)
  LAUNCH3(2, 1, 1, 5)   // (0,0,2)
  LAUNCH3(3, 1, 3, 3)   // (0,1,1)
  LAUNCH3(4, 1, 3, 5)   // (0,1,2)
  LAUNCH3(5, 1, 5, 5)   // (0,2,2)
  LAUNCH3(6, 3, 3, 3)   // (1,1,1)
  LAUNCH3(7, 3, 3, 5)   // (1,1,2)
  LAUNCH3(8, 3, 5, 5)   // (1,2,2)
  LAUNCH3(9, 5, 5, 5)   // (2,2,2)
#undef LAUNCH3
}